// SpikingNeuralNet_80857054315061
// MI455X (gfx1250) — compile-verified
//
#include <hip/hip_runtime.h>
#include <math.h>

// ---------------- problem constants (from reference) ----------------
#define NN 2048
#define TT 4096
#define MAXSPK 256
#define ALPHA_C 0.01f
#define DT_C (1.0f / 4096.0f)

// output layout (flat, return order): ys, tevents, yevents, event_types, num_spikes
constexpr size_t YS_ELEMS  = (size_t)TT * NN * 3;              // 25,165,824
constexpr size_t TEV_OFF   = YS_ELEMS;                          // +256
constexpr size_t YEV_OFF   = TEV_OFF + MAXSPK;                  // +1,572,864
constexpr size_t ET_OFF    = YEV_OFF + (size_t)MAXSPK * NN * 3; // +524,288
constexpr size_t NS_OFF    = ET_OFF + (size_t)MAXSPK * NN;      // +1

// ---------------- vector types ----------------
typedef int   v2i __attribute__((ext_vector_type(2)));
typedef float v2f __attribute__((ext_vector_type(2)));
typedef __attribute__((address_space(1))) v2i* g64p;  // global-AS pointer to 2xi32
typedef __attribute__((address_space(3))) v2i* l64p;  // LDS-AS pointer to 2xi32

// ---------------- CDNA5 async global->LDS helpers ----------------
#if __has_builtin(__builtin_amdgcn_global_load_async_to_lds_b64)
#define HAS_ASYNC_LDS 1
#else
#define HAS_ASYNC_LDS 0
#endif

__device__ __forceinline__ void async_copy_b64(const float* g, float* l) {
#if HAS_ASYNC_LDS
  __builtin_amdgcn_global_load_async_to_lds_b64(
      (g64p)(v2i*)g,   // global source (AS1)
      (l64p)(v2i*)l,   // LDS destination (AS3)
      /*offset=*/0, /*cpol=*/0);
#else
  ((v2f*)l)[0] = ((const v2f*)g)[0];
#endif
}

__device__ __forceinline__ void async_wait_all() {
#if HAS_ASYNC_LDS
#if __has_builtin(__builtin_amdgcn_s_wait_asynccnt)
  __builtin_amdgcn_s_wait_asynccnt(0);
#else
  asm volatile("s_wait_asynccnt 0" ::: "memory");
#endif
#endif
}

__device__ __forceinline__ void nt_store2(float a, float b, float* p) {
  v2f v; v.x = a; v.y = b;
  __builtin_nontemporal_store(v, (v2f*)p);   // global_store_b64 (non-temporal)
}

__device__ __forceinline__ float softplusf(float x) {
  // numerically stable softplus: max(x,0) + log1p(exp(-|x|))
  return fmaxf(x, 0.0f) + log1pf(expf(-fabsf(x)));
}

// ---------------- init kernel: default values for event outputs ----------------
__global__ void snn_init_out(float* __restrict__ out) {
  const size_t stride = (size_t)gridDim.x * blockDim.x;
  const size_t i = (size_t)blockIdx.x * blockDim.x + threadIdx.x;
  float* tev = out + TEV_OFF;
  float* yev = out + YEV_OFF;
  float* et  = out + ET_OFF;
  const float inf = __builtin_inff();
  if (i < (size_t)MAXSPK) tev[i] = inf;
  for (size_t k = i; k < (size_t)MAXSPK * NN * 3; k += stride) yev[k] = inf;
  for (size_t k = i; k < (size_t)MAXSPK * NN; k += stride) et[k] = 0.0f;
}

// ---------------- main persistent simulation kernel ----------------
// one workgroup = 1024 threads = 32 wave32s on one WGP; 2 neurons per thread
__global__ __launch_bounds__(1024, 1)
void snn_sim(const float* __restrict__ w, const float* __restrict__ mu,
             const float* __restrict__ v0, const float* __restrict__ i0,
             const float* __restrict__ ic, const float* __restrict__ u_init,
             const float* __restrict__ u_rs, float* __restrict__ out)
{
  __shared__ float icbuf[2][NN];   // 16 KB double buffer (async-filled)
  __shared__ float ubuf[2][NN];    // 16 KB double buffer (async-filled)
  __shared__ unsigned sh_min[2];   // ping-pong fused any/argmax reduction

  const int tid = threadIdx.x;
  const unsigned n0 = 2u * (unsigned)tid;
  const unsigned n1 = n0 + 1u;

  float* ys  = out;
  float* tev = out + TEV_OFF;
  float* yev = out + YEV_OFF;
  float* et  = out + ET_OFF;
  float* ns  = out + NS_OFF;

  const float mu1 = mu[0], mu2 = mu[1];
  const float a1    = DT_C * mu1;
  const float decay = 1.0f - DT_C * mu2;

  const v2f vv0 = *(const v2f*)(v0 + n0);
  const v2f ii0 = *(const v2f*)(i0 + n0);
  const v2f uu0 = *(const v2f*)(u_init + n0);
  float vx = vv0.x, vy = vv0.y;
  float ix = ii0.x, iy = ii0.y;
  float sx = logf(uu0.x) - ALPHA_C;
  float sy = logf(uu0.y) - ALPHA_C;

  if (tid == 0) { sh_min[0] = 0xFFFFFFFFu; sh_min[1] = 0xFFFFFFFFu; }

  // prologue: async-fill buffer 0 with row t=0
  async_copy_b64(ic   + n0, &icbuf[0][n0]);
  async_copy_b64(u_rs + n0, &ubuf[0][n0]);
  async_wait_all();
  __syncthreads();

  int cnt = 0;
  for (int t = 0; t < TT; ++t) {
    const int cur = t & 1;
    const int nxt = cur ^ 1;

    // kick off async prefetch of next step's ic/u rows into the other buffer
    if (t + 1 < TT) {
      const size_t roff = (size_t)(t + 1) * NN + n0;
      async_copy_b64(ic   + roff, &icbuf[nxt][n0]);
      async_copy_b64(u_rs + roff, &ubuf[nxt][n0]);
    }
    // progressively warm the 16MB weight matrix into L2 during early steps
    if (t < 512) {
      __builtin_prefetch(w + (size_t)t * 8192 + (size_t)tid * 8, 0, 0);
    }

    const float icx = icbuf[cur][n0], icy = icbuf[cur][n1];
    const float ux  = ubuf[cur][n0],  uy  = ubuf[cur][n1];

    // neuron dynamics
    const float v1x = fmaf(a1, ix + icx - vx, vx);
    const float v1y = fmaf(a1, iy + icy - vy, vy);
    const float i1x = ix * decay;
    const float i1y = iy * decay;
    const float s1x = fmaf(DT_C, softplusf(vx), sx);
    const float s1y = fmaf(DT_C, softplusf(vy), sy);
    const bool m0 = (s1x >= 0.0f);
    const bool m1 = (s1y >= 0.0f);

    // fused any()/argmax(): min masked neuron index via LDS atomic (ds_min_u32)
    if (m0 | m1) atomicMin(&sh_min[cur], m0 ? n0 : n1);
    __syncthreads();                              // B1: all mins visible
    const unsigned minidx = sh_min[cur];
    if (tid == 0) sh_min[nxt] = 0xFFFFFFFFu;      // prep next slot (pre-B2)
    const bool event = (minidx != 0xFFFFFFFFu);
    const bool wr    = event && (cnt < MAXSPK);

    if (wr) {
      const int idx = cnt;                        // cnt < MAXSPK guaranteed
      float* yrow = yev + ((size_t)idx * NN + n0) * 3;   // 8B-aligned (tid*24)
      nt_store2(v1x, i1x, yrow + 0);
      nt_store2(s1x, v1y, yrow + 2);
      nt_store2(i1y, s1y, yrow + 4);
      nt_store2(m0 ? 1.0f : 0.0f, m1 ? 1.0f : 0.0f, et + (size_t)idx * NN + n0);
      if (tid == 0) tev[idx] = (float)(t + 1) * DT_C;
    }

    // post-event transition
    float nvx = v1x, nvy = v1y, nix = i1x, niy = i1y, nsx = s1x, nsy = s1y;
    if (event) {
      const v2f wv = *(const v2f*)(w + (size_t)minidx * NN + n0);  // b64 gather
      nix = i1x + wv.x;
      niy = i1y + wv.y;
      if (m0) { nvx = v1x - 1.0f; nsx = logf(ux) - ALPHA_C; }
      if (m1) { nvy = v1y - 1.0f; nsy = logf(uy) - ALPHA_C; }
    }

    // stream out ys[t] (write-once data -> non-temporal b64s, keep L2 for w/ic/u)
    float* ysr = ys + ((size_t)t * NN + n0) * 3;       // 8B-aligned (tid*24)
    nt_store2(nvx, nix, ysr + 0);
    nt_store2(nsx, nvy, ysr + 2);
    nt_store2(niy, nsy, ysr + 4);

    vx = nvx; vy = nvy; ix = nix; iy = niy; sx = nsx; sy = nsy;
    cnt += wr ? 1 : 0;

    // make sure next buffer's async fill landed before anyone reads it
    async_wait_all();
    __syncthreads();                              // B2
  }

  if (tid == 0) ns[0] = (float)cnt;
}

// ---------------- launch ----------------
extern "C" void kernel_launch(void* const* d_in, const int* in_sizes, int n_in,
                              void* d_out, int out_size, void* d_ws, size_t ws_size,
                              hipStream_t stream) {
  const float* w  = (const float*)d_in[0];
  const float* mu = (const float*)d_in[1];
  const float* v0 = (const float*)d_in[2];
  const float* i0 = (const float*)d_in[3];
  const float* ic = (const float*)d_in[4];
  const float* ui = (const float*)d_in[5];
  const float* ur = (const float*)d_in[6];
  float* out = (float*)d_out;

  snn_init_out<<<1024, 256, 0, stream>>>(out);
  snn_sim<<<1, 1024, 0, stream>>>(w, mu, v0, i0, ic, ui, ur, out);
}